// PointTransformerLayer_23287312679061
// MI455X (gfx1250) — compile-verified
//
#include <hip/hip_runtime.h>
#include <hip/hip_bf16.h>
#include <math.h>

// ---------------------------------------------------------------------------
// PointTransformer layer for MI455X (gfx1250, wave32, WMMA).
// All big intermediates (xq/xk/xv 61MB, pr1 7.7MB, h1 41MB) are L2-resident
// (192MB L2); gathered (N,K,C) tensors are never materialized.
// f32 WMMA (v_wmma_f32_16x16x4_f32) used for the QKV GEMMs and the fused
// BN->ReLU->(128x16) GEMM.
// ---------------------------------------------------------------------------

#define NPTS 40000
#define CCH  128
#define KNB  16
#define CS   16
#define MROW (NPTS * KNB)     // 640000
#define EPSV 1e-5f

typedef __attribute__((ext_vector_type(2))) float v2f;
typedef __attribute__((ext_vector_type(8))) float v8f;

// workspace layout (floats)
#define OFF_XQ    ((size_t)0)
#define OFF_XK    (OFF_XQ + (size_t)NPTS * CCH)
#define OFF_XV    (OFF_XK + (size_t)NPTS * CCH)
#define OFF_PR1   (OFF_XV + (size_t)NPTS * CCH)
#define OFF_H1    (OFF_PR1 + (size_t)MROW * 3)
#define OFF_STATS (OFF_H1 + (size_t)MROW * CS)
// stats region: sum3[3] sq3[3] scale3[3] shift3[3]
//               sumC[128] sqC[128] scaleC[128] shiftC[128]
//               sum16[16] sq16[16] scale16[16] shift16[16]
#define ST_SUM3    0
#define ST_SQ3     3
#define ST_SCALE3  6
#define ST_SHIFT3  9
#define ST_SUMC    12
#define ST_SQC     (ST_SUMC + 128)
#define ST_SCALEC  (ST_SQC + 128)
#define ST_SHIFTC  (ST_SCALEC + 128)
#define ST_SUM16   (ST_SHIFTC + 128)
#define ST_SQ16    (ST_SUM16 + 16)
#define ST_SCALE16 (ST_SQ16 + 16)
#define ST_SHIFT16 (ST_SCALE16 + 16)
#define ST_TOTAL   (ST_SHIFT16 + 16)   // 588 floats

__global__ void zero_stats_kernel(float* st, int n) {
    int i = blockIdx.x * blockDim.x + threadIdx.x;
    if (i < n) st[i] = 0.f;
}

// ---------------------------------------------------------------------------
// Kernel 1: xq = x@wq+bq, xk = x@wk+bk, xv = x@wv+bv   (WMMA f32 16x16x4)
// 256 threads = 8 waves; block owns a 16-row tile of x (staged in LDS,
// padded stride 132 to avoid bank conflicts); wave w owns column tile w.
// ---------------------------------------------------------------------------
__global__ __launch_bounds__(256) void qkv_gemm_kernel(
    const float* __restrict__ x,
    const float* __restrict__ wq, const float* __restrict__ bq,
    const float* __restrict__ wk, const float* __restrict__ bk,
    const float* __restrict__ wv, const float* __restrict__ bv,
    float* __restrict__ xq, float* __restrict__ xk, float* __restrict__ xv)
{
    __shared__ float As[16 * 132];
    const int tid = threadIdx.x;
    const int rowBase = blockIdx.x * 16;

    for (int e = tid; e < 16 * 128; e += 256) {
        int r = e >> 7, c = e & 127;
        As[r * 132 + c] = x[(size_t)(rowBase + r) * CCH + c];
    }
    __syncthreads();

    const int wave = tid >> 5;
    const int lane = tid & 31;
    const int hh   = lane >> 4;   // half select
    const int l15  = lane & 15;
    const int colBase = wave * 16;

    const float* Wm[3] = {wq, wk, wv};
    const float* Bm[3] = {bq, bk, bv};
    float*       Om[3] = {xq, xk, xv};

#pragma unroll
    for (int z = 0; z < 3; ++z) {
        const float* W = Wm[z];
        v8f acc = {};
        for (int kk = 0; kk < 128; kk += 4) {
            v2f a, b;
            // A 16x4: lanes 0-15 hold K=kk..kk+1, lanes 16-31 hold K=kk+2..kk+3
            a.x = As[l15 * 132 + kk + 2 * hh + 0];
            a.y = As[l15 * 132 + kk + 2 * hh + 1];
            // B 4x16: mirrored K striping, row-major W[k][col]
            b.x = W[(size_t)(kk + 2 * hh + 0) * CCH + colBase + l15];
            b.y = W[(size_t)(kk + 2 * hh + 1) * CCH + colBase + l15];
            acc = __builtin_amdgcn_wmma_f32_16x16x4_f32(
                false, a, false, b, (short)0, acc, false, false);
        }
        const float bias = Bm[z][colBase + l15];
        float* O = Om[z];
#pragma unroll
        for (int r = 0; r < 8; ++r) {
            int row = rowBase + r + hh * 8;
            O[(size_t)row * CCH + colBase + l15] = acc[r] + bias;
        }
    }
}

// ---------------------------------------------------------------------------
// Kernel 2: pr1 = (p[idx]-p) @ p_w1 + p_b1  (N*K rows, 3 ch) + BN(3) stats
// ---------------------------------------------------------------------------
__global__ __launch_bounds__(256) void pr1_stats_kernel(
    const float* __restrict__ p, const int* __restrict__ idx,
    const float* __restrict__ p_w1, const float* __restrict__ p_b1,
    float* __restrict__ pr1, float* __restrict__ st)
{
    __shared__ float s[6];
    if (threadIdx.x < 6) s[threadIdx.x] = 0.f;
    __syncthreads();

    float ls0 = 0, ls1 = 0, ls2 = 0, lq0 = 0, lq1 = 0, lq2 = 0;
    for (int row = blockIdx.x * blockDim.x + threadIdx.x; row < MROW;
         row += gridDim.x * blockDim.x) {
        int n  = row >> 4;
        int pi = idx[row];
        float dx = p[pi * 3 + 0] - p[n * 3 + 0];
        float dy = p[pi * 3 + 1] - p[n * 3 + 1];
        float dz = p[pi * 3 + 2] - p[n * 3 + 2];
        float t0 = dx * p_w1[0] + dy * p_w1[3] + dz * p_w1[6] + p_b1[0];
        float t1 = dx * p_w1[1] + dy * p_w1[4] + dz * p_w1[7] + p_b1[1];
        float t2 = dx * p_w1[2] + dy * p_w1[5] + dz * p_w1[8] + p_b1[2];
        pr1[(size_t)row * 3 + 0] = t0;
        pr1[(size_t)row * 3 + 1] = t1;
        pr1[(size_t)row * 3 + 2] = t2;
        ls0 += t0; ls1 += t1; ls2 += t2;
        lq0 += t0 * t0; lq1 += t1 * t1; lq2 += t2 * t2;
    }
    atomicAdd(&s[0], ls0); atomicAdd(&s[1], ls1); atomicAdd(&s[2], ls2);
    atomicAdd(&s[3], lq0); atomicAdd(&s[4], lq1); atomicAdd(&s[5], lq2);
    __syncthreads();
    if (threadIdx.x < 3) {
        atomicAdd(&st[ST_SUM3 + threadIdx.x], s[threadIdx.x]);
        atomicAdd(&st[ST_SQ3  + threadIdx.x], s[3 + threadIdx.x]);
    }
}

// ---------------------------------------------------------------------------
// Finalize BN: scale = g*rsqrt(var+eps), shift = be - mean*scale
// ---------------------------------------------------------------------------
__global__ void finalize_bn_kernel(
    const float* __restrict__ sum, const float* __restrict__ sq,
    const float* __restrict__ g, const float* __restrict__ be,
    float* __restrict__ scale, float* __restrict__ shift, int nch, float invM)
{
    int c = blockIdx.x * blockDim.x + threadIdx.x;
    if (c >= nch) return;
    float m  = sum[c] * invM;
    float v  = sq[c] * invM - m * m;
    float sc = g[c] * rsqrtf(v + EPSV);
    scale[c] = sc;
    shift[c] = be[c] - m * sc;
}

// ---------------------------------------------------------------------------
// Kernel 4: BN(128) stats over w_pre = xk[idx]-xq + linear_p(pr1)
// (w_pre recomputed here and in kernel 6 rather than materializing 327MB)
// blockDim=128: thread = channel, block strides over rows.
// ---------------------------------------------------------------------------
__global__ __launch_bounds__(128) void wpre_stats_kernel(
    const float* __restrict__ xq, const float* __restrict__ xk,
    const int* __restrict__ idx, const float* __restrict__ pr1,
    const float* __restrict__ st_in,  // scale3/shift3 live here
    const float* __restrict__ p_w2, const float* __restrict__ p_b2,
    float* __restrict__ st)
{
    const int c = threadIdx.x;
    const float s30 = st_in[ST_SCALE3 + 0], s31 = st_in[ST_SCALE3 + 1],
                s32 = st_in[ST_SCALE3 + 2];
    const float h30 = st_in[ST_SHIFT3 + 0], h31 = st_in[ST_SHIFT3 + 1],
                h32 = st_in[ST_SHIFT3 + 2];
    const float w0 = p_w2[c], w1 = p_w2[CCH + c], w2 = p_w2[2 * CCH + c];
    const float wb = p_b2[c];

    float ls = 0.f, lq = 0.f;
    for (int row = blockIdx.x; row < MROW; row += gridDim.x) {
        int n  = row >> 4;
        int pi = idx[row];
        float q0 = fmaxf(pr1[(size_t)row * 3 + 0] * s30 + h30, 0.f);
        float q1 = fmaxf(pr1[(size_t)row * 3 + 1] * s31 + h31, 0.f);
        float q2 = fmaxf(pr1[(size_t)row * 3 + 2] * s32 + h32, 0.f);
        float w = xk[(size_t)pi * CCH + c] - xq[(size_t)n * CCH + c]
                + q0 * w0 + q1 * w1 + q2 * w2 + wb;
        ls += w; lq += w * w;
    }
    atomicAdd(&st[ST_SUMC + c], ls);
    atomicAdd(&st[ST_SQC  + c], lq);
}

// ---------------------------------------------------------------------------
// Kernel 6: h1 = relu(BN128(w_pre)) @ w_w1 + w_b1   (WMMA f32 16x16x4)
// blockDim=128 (4 waves); block stages a 64x128 activation tile in LDS
// (padded stride 132), each wave WMMA-multiplies one 16-row tile by w_w1.
// ---------------------------------------------------------------------------
__global__ __launch_bounds__(128) void h1_gemm_kernel(
    const float* __restrict__ xq, const float* __restrict__ xk,
    const int* __restrict__ idx, const float* __restrict__ pr1,
    const float* __restrict__ st,
    const float* __restrict__ p_w2, const float* __restrict__ p_b2,
    const float* __restrict__ w_w1, const float* __restrict__ w_b1,
    float* __restrict__ h1)
{
    __shared__ float As[64 * 132];
    const int tid = threadIdx.x;
    const int rowBase = blockIdx.x * 64;

    {
        const int c = tid;  // 128 threads = 128 channels
        const float s30 = st[ST_SCALE3 + 0], s31 = st[ST_SCALE3 + 1],
                    s32 = st[ST_SCALE3 + 2];
        const float h30 = st[ST_SHIFT3 + 0], h31 = st[ST_SHIFT3 + 1],
                    h32 = st[ST_SHIFT3 + 2];
        const float w0 = p_w2[c], w1 = p_w2[CCH + c], w2 = p_w2[2 * CCH + c];
        const float wb = p_b2[c];
        const float sc = st[ST_SCALEC + c], sh = st[ST_SHIFTC + c];
        for (int r = 0; r < 64; ++r) {
            int row = rowBase + r;
            int n   = row >> 4;
            int pi  = idx[row];
            float q0 = fmaxf(pr1[(size_t)row * 3 + 0] * s30 + h30, 0.f);
            float q1 = fmaxf(pr1[(size_t)row * 3 + 1] * s31 + h31, 0.f);
            float q2 = fmaxf(pr1[(size_t)row * 3 + 2] * s32 + h32, 0.f);
            float w = xk[(size_t)pi * CCH + c] - xq[(size_t)n * CCH + c]
                    + q0 * w0 + q1 * w1 + q2 * w2 + wb;
            As[r * 132 + c] = fmaxf(w * sc + sh, 0.f);
        }
    }
    __syncthreads();

    const int wave = tid >> 5;
    const int lane = tid & 31;
    const int hh   = lane >> 4;
    const int l15  = lane & 15;
    const int r0   = wave * 16;

    v8f acc = {};
    for (int kk = 0; kk < 128; kk += 4) {
        v2f a, b;
        a.x = As[(r0 + l15) * 132 + kk + 2 * hh + 0];
        a.y = As[(r0 + l15) * 132 + kk + 2 * hh + 1];
        b.x = w_w1[(kk + 2 * hh + 0) * CS + l15];
        b.y = w_w1[(kk + 2 * hh + 1) * CS + l15];
        acc = __builtin_amdgcn_wmma_f32_16x16x4_f32(
            false, a, false, b, (short)0, acc, false, false);
    }
    const float bias = w_b1[l15];
#pragma unroll
    for (int r = 0; r < 8; ++r) {
        int grow = rowBase + r0 + r + hh * 8;
        h1[(size_t)grow * CS + l15] = acc[r] + bias;
    }
}

// ---------------------------------------------------------------------------
// Kernel 6b: BN(16) stats over h1
// ---------------------------------------------------------------------------
__global__ __launch_bounds__(256) void stats16_kernel(
    const float* __restrict__ h1, float* __restrict__ st)
{
    __shared__ float s[32];
    if (threadIdx.x < 32) s[threadIdx.x] = 0.f;
    __syncthreads();
    const int c = threadIdx.x & 15;  // stride (gridDim*256) is a mult of 16
    float ls = 0.f, lq = 0.f;
    for (size_t e = (size_t)blockIdx.x * 256 + threadIdx.x;
         e < (size_t)MROW * CS; e += (size_t)gridDim.x * 256) {
        float v = h1[e];
        ls += v; lq += v * v;
    }
    atomicAdd(&s[c], ls);
    atomicAdd(&s[16 + c], lq);
    __syncthreads();
    if (threadIdx.x < 16) {
        atomicAdd(&st[ST_SUM16 + threadIdx.x], s[threadIdx.x]);
        atomicAdd(&st[ST_SQ16  + threadIdx.x], s[16 + threadIdx.x]);
    }
}

// ---------------------------------------------------------------------------
// Kernel 8: per-point fused tail. One wave32 per point.
// lane = (h = lane>>4, j = lane&15): lane handles attention column j and the
// 8 neighbors {h, h+2, ..., h+14}; the two halves combine via shfl_xor(16).
//   h2 = relu(BN16(h1)) @ w_w2 + w_b2 -> softmax over K (per column) ->
//   out[n, s*16+j] = sum_k wgt[k][j] * (xv[idx]+linear_p(pr1))[s*16+j]
// ---------------------------------------------------------------------------
__global__ __launch_bounds__(256) void attn_out_kernel(
    const float* __restrict__ xv, const int* __restrict__ idx,
    const float* __restrict__ pr1, const float* __restrict__ st,
    const float* __restrict__ p_w2, const float* __restrict__ p_b2,
    const float* __restrict__ h1,
    const float* __restrict__ w_w2, const float* __restrict__ w_b2,
    float* __restrict__ out)
{
    const int n    = blockIdx.x * 8 + (threadIdx.x >> 5);
    const int lane = threadIdx.x & 31;
    const int j    = lane & 15;
    const int hf   = lane >> 4;

    float wcol[16], sc16[16], sh16[16];
#pragma unroll
    for (int jj = 0; jj < 16; ++jj) {
        wcol[jj] = w_w2[jj * 16 + j];
        sc16[jj] = st[ST_SCALE16 + jj];
        sh16[jj] = st[ST_SHIFT16 + jj];
    }
    const float b2 = w_b2[j];

    // h2 logits for my 8 neighbors
    float hk[8];
#pragma unroll
    for (int t = 0; t < 8; ++t) {
        int k = hf + 2 * t;
        const float* hr = h1 + ((size_t)n * KNB + k) * CS;
        float a = b2;
#pragma unroll
        for (int jj = 0; jj < 16; ++jj)
            a += fmaxf(hr[jj] * sc16[jj] + sh16[jj], 0.f) * wcol[jj];
        hk[t] = a;
    }
    // softmax over K (16 neighbors) per column j
    float mx = hk[0];
#pragma unroll
    for (int t = 1; t < 8; ++t) mx = fmaxf(mx, hk[t]);
    mx = fmaxf(mx, __shfl_xor(mx, 16));
    float ex[8];
    float se = 0.f;
#pragma unroll
    for (int t = 0; t < 8; ++t) { ex[t] = __expf(hk[t] - mx); se += ex[t]; }
    se += __shfl_xor(se, 16);
    const float inv = 1.f / se;

    const float s30 = st[ST_SCALE3 + 0], s31 = st[ST_SCALE3 + 1],
                s32 = st[ST_SCALE3 + 2];
    const float h30 = st[ST_SHIFT3 + 0], h31 = st[ST_SHIFT3 + 1],
                h32 = st[ST_SHIFT3 + 2];

    float acc[8] = {0, 0, 0, 0, 0, 0, 0, 0};
#pragma unroll
    for (int t = 0; t < 8; ++t) {
        int k   = hf + 2 * t;
        int row = n * KNB + k;
        int pi  = idx[row];
        float q0 = fmaxf(pr1[(size_t)row * 3 + 0] * s30 + h30, 0.f);
        float q1 = fmaxf(pr1[(size_t)row * 3 + 1] * s31 + h31, 0.f);
        float q2 = fmaxf(pr1[(size_t)row * 3 + 2] * s32 + h32, 0.f);
        float wg = ex[t] * inv;
#pragma unroll
        for (int s = 0; s < 8; ++s) {
            int c = s * 16 + j;
            float val = xv[(size_t)pi * CCH + c]
                      + q0 * p_w2[c] + q1 * p_w2[CCH + c] + q2 * p_w2[2 * CCH + c]
                      + p_b2[c];
            acc[s] += wg * val;
        }
    }
#pragma unroll
    for (int s = 0; s < 8; ++s) {
        float tot = acc[s] + __shfl_xor(acc[s], 16);
        if (hf == 0) out[(size_t)n * CCH + s * 16 + j] = tot;
    }
}

// ---------------------------------------------------------------------------
extern "C" void kernel_launch(void* const* d_in, const int* in_sizes, int n_in,
                              void* d_out, int out_size, void* d_ws, size_t ws_size,
                              hipStream_t stream)
{
    const float* p    = (const float*)d_in[0];
    const float* x    = (const float*)d_in[1];
    const int*   idx  = (const int*)  d_in[2];
    const float* wq   = (const float*)d_in[3];
    const float* bq   = (const float*)d_in[4];
    const float* wk   = (const float*)d_in[5];
    const float* bk   = (const float*)d_in[6];
    const float* wv   = (const float*)d_in[7];
    const float* bv   = (const float*)d_in[8];
    const float* p_w1 = (const float*)d_in[9];
    const float* p_b1 = (const float*)d_in[10];
    const float* p_g1 = (const float*)d_in[11];
    const float* p_be1= (const float*)d_in[12];
    const float* p_w2 = (const float*)d_in[13];
    const float* p_b2 = (const float*)d_in[14];
    const float* w_g1 = (const float*)d_in[15];
    const float* w_be1= (const float*)d_in[16];
    const float* w_w1 = (const float*)d_in[17];
    const float* w_b1 = (const float*)d_in[18];
    const float* w_g2 = (const float*)d_in[19];
    const float* w_be2= (const float*)d_in[20];
    const float* w_w2 = (const float*)d_in[21];
    const float* w_b2 = (const float*)d_in[22];

    float* ws  = (float*)d_ws;
    float* xq  = ws + OFF_XQ;
    float* xk  = ws + OFF_XK;
    float* xv  = ws + OFF_XV;
    float* pr1 = ws + OFF_PR1;
    float* h1  = ws + OFF_H1;
    float* st  = ws + OFF_STATS;

    const float invM = 1.0f / (float)MROW;

    // 0) zero stats accumulators (harness does not re-poison between replays)
    zero_stats_kernel<<<(ST_TOTAL + 255) / 256, 256, 0, stream>>>(st, ST_TOTAL);

    // 1) QKV projections (WMMA)
    qkv_gemm_kernel<<<NPTS / 16, 256, 0, stream>>>(
        x, wq, bq, wk, bk, wv, bv, xq, xk, xv);

    // 2) positional pre-activation + BN(3) stats
    pr1_stats_kernel<<<1024, 256, 0, stream>>>(p, idx, p_w1, p_b1, pr1, st);
    finalize_bn_kernel<<<1, 32, 0, stream>>>(
        st + ST_SUM3, st + ST_SQ3, p_g1, p_be1,
        st + ST_SCALE3, st + ST_SHIFT3, 3, invM);

    // 3) BN(128) stats over w_pre (recomputed, never materialized)
    wpre_stats_kernel<<<1024, 128, 0, stream>>>(
        xq, xk, idx, pr1, st, p_w2, p_b2, st);
    finalize_bn_kernel<<<1, 128, 0, stream>>>(
        st + ST_SUMC, st + ST_SQC, w_g1, w_be1,
        st + ST_SCALEC, st + ST_SHIFTC, CCH, invM);

    // 4) h1 = relu(BN(w_pre)) @ w_w1 + b (WMMA), then BN(16) stats
    h1_gemm_kernel<<<MROW / 64, 128, 0, stream>>>(
        xq, xk, idx, pr1, st, p_w2, p_b2, w_w1, w_b1, h1);
    stats16_kernel<<<512, 256, 0, stream>>>(h1, st);
    finalize_bn_kernel<<<1, 32, 0, stream>>>(
        st + ST_SUM16, st + ST_SQ16, w_g2, w_be2,
        st + ST_SCALE16, st + ST_SHIFT16, CS, invM);

    // 5) fused h2 / softmax / share-plane weighted sum
    attn_out_kernel<<<NPTS / 8, 256, 0, stream>>>(
        xv, idx, pr1, st, p_w2, p_b2, h1, w_w2, w_b2, (float*)d_out);
}